// SequenceMaskingNet_35605278884050
// MI455X (gfx1250) — compile-verified
//
#include <hip/hip_runtime.h>
#include <hip/hip_bf16.h>
#include <stdint.h>

// Problem: x [T=2048, B=32, D=512] fp32, W [512,512], b [512]
// Y = X @ W^T + b  (M = T*B = 65536, N = 512, K = 512), then bidirectional LIF.

typedef __attribute__((ext_vector_type(2))) float v2f;
typedef __attribute__((ext_vector_type(8))) float v8f;

#define MDIM 65536
#define NDIM 512
#define KDIM 512
#define BM 128
#define BN 128
#define BK 32
#define LDK (BK + 4)   // row stride 36 floats: float4-aligned, conflict-free frag reads

#if __has_builtin(__builtin_amdgcn_global_load_async_to_lds_b128)
#define HAVE_ASYNC 1
#else
#define HAVE_ASYNC 0
#endif

__device__ __forceinline__ void wait_async_all() {
#if __has_builtin(__builtin_amdgcn_s_wait_asynccnt)
  __builtin_amdgcn_s_wait_asynccnt(0);
#else
  asm volatile("s_wait_asynccnt 0x0" ::: "memory");
#endif
}

#if HAVE_ASYNC
// builtin signature (from hipcc diagnostic): param0 = AS1 ptr to int __vector(4)
typedef int v4i_vec __attribute__((vector_size(16)));
typedef __attribute__((address_space(1))) v4i_vec v4i_glob;
typedef __attribute__((address_space(3))) v4i_vec v4i_lds;

__device__ __forceinline__ void async_b128(const void* gsrc, void* ldst) {
  // memory -> LDS, 16B per lane, tracked by ASYNCcnt
  __builtin_amdgcn_global_load_async_to_lds_b128(
      (v4i_glob*)(uint64_t)(uintptr_t)gsrc,
      (v4i_lds*)(uint32_t)(uintptr_t)ldst, 0, 0);
}
#endif

// ---------------------------------------------------------------------------
// GEMM: Y[m, n] = sum_k X[m, k] * W[n, k] + bias[n]   (fp32, WMMA 16x16x4)
// 256 threads = 8 waves (4Mx2N); wave tile 32x64 = 2x4 v8f accumulators.
// Both tiles kept in natural [row][k] layout -> every fragment is one b64 read.
// Double-buffered async-to-LDS pipeline.
// ---------------------------------------------------------------------------
__global__ __launch_bounds__(256) void gemm_bias_kernel(
    const float* __restrict__ X, const float* __restrict__ W,
    const float* __restrict__ bias, float* __restrict__ Y) {
  __shared__ float As[2][BM][LDK];   // X tile  [m][k]
  __shared__ float Ws[2][BN][LDK];   // W tile  [n][k]  (natural layout!)

  const int tid   = threadIdx.x;
  const int lane  = tid & 31;
  const int wave  = tid >> 5;
  const int waveM = wave >> 1;       // 0..3
  const int waveN = wave & 1;        // 0..1
  const int tileM = blockIdx.y * BM;
  const int tileN = blockIdx.x * BN;

  v8f acc[2][4];
  #pragma unroll
  for (int mi = 0; mi < 2; ++mi)
    #pragma unroll
    for (int nf = 0; nf < 4; ++nf) {
      v8f z = {};
      acc[mi][nf] = z;
    }

  // cooperative loads: 256 threads x 4 float4 per 128x32 tile
  const int ldr = tid >> 3;          // 0..31
  const int ldc = (tid & 7) * 4;     // 0,4,...,28

  // fragment coordinates (wave32 WMMA VGPR layouts)
  const int amRow = waveM * 32 + (lane & 15);   // + mi*16
  const int akOff = (lane >> 4) * 2;            // lanes 0-15: K+0/1; 16-31: K+2/3
  const int bnCol = waveN * 64 + (lane & 15);   // + nf*16
  const int bkOff = (lane >> 4) * 2;

  auto load_tile = [&](int buf, int k0) {
    #pragma unroll
    for (int i = 0; i < 4; ++i) {
      const int r = ldr + i * 32;
      const float* xs = X + (size_t)(tileM + r) * KDIM + (k0 + ldc);
      const float* ws = W + (size_t)(tileN + r) * KDIM + (k0 + ldc);
#if HAVE_ASYNC
      async_b128(xs, &As[buf][r][ldc]);
      async_b128(ws, &Ws[buf][r][ldc]);
#else
      *(float4*)&As[buf][r][ldc] = *(const float4*)xs;
      *(float4*)&Ws[buf][r][ldc] = *(const float4*)ws;
#endif
    }
  };

  load_tile(0, 0);
  int p = 0;
  for (int k0 = 0; k0 < KDIM; k0 += BK) {
#if HAVE_ASYNC
    wait_async_all();                 // buffer p resident in LDS
#endif
    __syncthreads();                  // ...and all waves done reading buffer p^1
    if (k0 + BK < KDIM) load_tile(p ^ 1, k0 + BK);   // overlap with compute

    const float (*Ap)[LDK] = As[p];
    const float (*Wp)[LDK] = Ws[p];
    #pragma unroll
    for (int kq = 0; kq < BK; kq += 4) {
      v2f a[2], bf[4];
      #pragma unroll
      for (int mi = 0; mi < 2; ++mi)
        a[mi] = *(const v2f*)&Ap[amRow + mi * 16][kq + akOff];
      #pragma unroll
      for (int nf = 0; nf < 4; ++nf)
        bf[nf] = *(const v2f*)&Wp[bnCol + nf * 16][kq + bkOff];
      #pragma unroll
      for (int mi = 0; mi < 2; ++mi)
        #pragma unroll
        for (int nf = 0; nf < 4; ++nf)
          acc[mi][nf] = __builtin_amdgcn_wmma_f32_16x16x4_f32(
              false, a[mi], false, bf[nf], (short)0, acc[mi][nf], false, false);
    }
    p ^= 1;
  }

  // epilogue: C/D layout -> VGPR r holds row r (lanes 0-15) / r+8 (lanes 16-31)
  const int nCol   = lane & 15;
  const int rowOff = (lane >> 4) * 8;
  #pragma unroll
  for (int mi = 0; mi < 2; ++mi) {
    const int mbase = tileM + waveM * 32 + mi * 16 + rowOff;
    #pragma unroll
    for (int nf = 0; nf < 4; ++nf) {
      const int n = tileN + waveN * 64 + nf * 16 + nCol;
      const float bb = bias[n];
      #pragma unroll
      for (int r = 0; r < 8; ++r)
        Y[(size_t)(mbase + r) * NDIM + n] = acc[mi][nf][r] + bb;
    }
  }
}

// ---------------------------------------------------------------------------
// Bidirectional LIF scan; one thread per (b,d) column. Forward spikes are
// bit-packed into LDS (64 words / thread), then the backward-in-time scan
// emits out[t] = s_bwd + fwd_bit[t]. Y read twice, out written once.
// ---------------------------------------------------------------------------
#define TSTEPS 2048
#define NCOLS 16384
#define SCAN_BLK 128

__global__ __launch_bounds__(SCAN_BLK) void lif_bidir_kernel(
    const float* __restrict__ Y, float* __restrict__ out) {
  __shared__ unsigned int sb[64 * SCAN_BLK];  // 32 KB spike bitmap
  const int tid = threadIdx.x;
  const int col = blockIdx.x * SCAN_BLK + tid;

  float v = 0.f;
  for (int w = 0; w < 64; ++w) {
    unsigned int m = 0u;
    #pragma unroll
    for (int i = 0; i < 32; ++i) {
      const int t = w * 32 + i;
      if (t + 32 < TSTEPS)
        __builtin_prefetch(&Y[(size_t)(t + 32) * NCOLS + col], 0, 0);
      const float c = Y[(size_t)t * NCOLS + col];
      v = 0.5f * (v + c);
      if (v >= 1.0f) { m |= (1u << i); v = 0.f; }
    }
    sb[w * SCAN_BLK + tid] = m;
  }

  v = 0.f;
  for (int w = 63; w >= 0; --w) {
    const unsigned int m = sb[w * SCAN_BLK + tid];
    #pragma unroll
    for (int i = 31; i >= 0; --i) {
      const int t = w * 32 + i;
      if (t >= 32)
        __builtin_prefetch(&Y[(size_t)(t - 32) * NCOLS + col], 0, 0);
      const float c = Y[(size_t)t * NCOLS + col];
      v = 0.5f * (v + c);
      float s = 0.f;
      if (v >= 1.0f) { s = 1.f; v = 0.f; }
      out[(size_t)t * NCOLS + col] = s + (float)((m >> i) & 1u);
    }
  }
}

extern "C" void kernel_launch(void* const* d_in, const int* in_sizes, int n_in,
                              void* d_out, int out_size, void* d_ws, size_t ws_size,
                              hipStream_t stream) {
  const float* X    = (const float*)d_in[0];  // [2048, 32, 512]
  const float* W    = (const float*)d_in[1];  // [512, 512]
  const float* bias = (const float*)d_in[2];  // [512]
  float* out = (float*)d_out;                 // [2048, 32, 512]
  float* Y   = (float*)d_ws;                  // 65536 x 512 fp32 = 128 MB scratch

  dim3 gemmGrid(NDIM / BN, MDIM / BM, 1);     // 4 x 512 blocks
  gemm_bias_kernel<<<gemmGrid, 256, 0, stream>>>(X, W, bias, Y);

  lif_bidir_kernel<<<NCOLS / SCAN_BLK, SCAN_BLK, 0, stream>>>(Y, out);
}